// GraphConvSparse_85804856639952
// MI455X (gfx1250) — compile-verified
//
#include <hip/hip_runtime.h>

typedef float v2f __attribute__((ext_vector_type(2)));
typedef float v8f __attribute__((ext_vector_type(8)));

#define IN_DIM  128
#define OUT_DIM 64

// ---------------------------------------------------------------------------
// Kernel 1: xw[N,64] = x[N,128] @ W[128,64] via V_WMMA_F32_16X16X4_F32.
// 8 waves/block, each wave computes a 16x64 tile (4 accumulators of 16x16).
// W is staged in LDS pre-packed as k-pairs so each B fragment is a single
// aligned ds_load_b64 (no VGPR repacking movs before the WMMAs).
// ---------------------------------------------------------------------------
__global__ __launch_bounds__(256) void gemm_xw_kernel(const float* __restrict__ x,
                                                      const float* __restrict__ W,
                                                      float* __restrict__ xw,
                                                      int N) {
    // Wp[p][c] = (W[2p][c], W[2p+1][c]), p = 0..63, c = 0..63  -> 32 KB
    __shared__ float2 Wp[(IN_DIM / 2) * OUT_DIM];

    const int tid = threadIdx.x;

    // Cooperative pack: 4096 float2 entries / 256 threads = 16 each.
    #pragma unroll
    for (int i = 0; i < (IN_DIM / 2) * OUT_DIM / 256; ++i) {
        const int idx = i * 256 + tid;
        const int p = idx >> 6;          // k-pair index
        const int c = idx & 63;          // column
        float2 v;
        v.x = W[(2 * p)     * OUT_DIM + c];
        v.y = W[(2 * p + 1) * OUT_DIM + c];
        Wp[idx] = v;
    }
    __syncthreads();

    const int lane  = tid & 31;
    const int wave  = tid >> 5;
    const int row_base = (blockIdx.x * 8 + wave) * 16;
    if (row_base >= N) return;          // wave-uniform: EXEC stays all-ones below

    const int m     = lane & 15;        // A: row-in-tile; B/C/D: col-in-tile
    const int khalf = lane >> 4;        // 0 -> K=0,1 ; 1 -> K=2,3  (ISA A layout)

    const float* xrow = x + (size_t)(row_base + m) * IN_DIM;
    const float2* __restrict__ Wrow = Wp + m;   // + t*16 + pair*64

    v8f acc[4];
    acc[0] = v8f{}; acc[1] = v8f{}; acc[2] = v8f{}; acc[3] = v8f{};

    #pragma unroll 4
    for (int k = 0; k < IN_DIM; k += 4) {
        const int kk   = k + khalf * 2;      // this lane's K offset
        const int pair = (k >> 1) + khalf;   // kk/2
        // A fragment: 8B-aligned global_load_b64 (kk even)
        v2f a = *(const v2f*)(xrow + kk);
        #pragma unroll
        for (int t = 0; t < 4; ++t) {
            // B fragment: single aligned ds_load_b64
            v2f b = *(const v2f*)(Wrow + pair * OUT_DIM + t * 16);
            // 8 args: (neg_a, A, neg_b, B, c_mod, C, reuse_a, reuse_b)
            acc[t] = __builtin_amdgcn_wmma_f32_16x16x4_f32(
                false, a, false, b, (short)0, acc[t], false, false);
        }
    }

    // C/D layout: VGPR j holds row M=j (lanes 0-15) / M=8+j (lanes 16-31), col N=m.
    const int rofs = khalf * 8;
    float* orow = xw + (size_t)(row_base + rofs) * OUT_DIM + m;
    #pragma unroll
    for (int j = 0; j < 8; ++j) {
        const size_t o = (size_t)j * OUT_DIM;
        orow[o +  0] = acc[0][j];
        orow[o + 16] = acc[1][j];
        orow[o + 32] = acc[2][j];
        orow[o + 48] = acc[3][j];
    }
}

// ---------------------------------------------------------------------------
// Kernel 2: zero the accumulation buffer (128-bit stores).
// ---------------------------------------------------------------------------
__global__ __launch_bounds__(256) void zero_kernel(float4* __restrict__ out, int n4) {
    const int i = blockIdx.x * 256 + threadIdx.x;
    if (i < n4) out[i] = make_float4(0.f, 0.f, 0.f, 0.f);
}

// ---------------------------------------------------------------------------
// Kernel 3: weighted scatter: out[dst] += val * xw[src].
// 16 threads per edge; each gathers a float4 of xw (global_load_b128) and
// issues 4 global_atomic_add_f32. xw (25.6 MB) and out (25.6 MB) are
// L2-resident (192 MB L2), so gathers and atomics stay on-chip.
// ---------------------------------------------------------------------------
__global__ __launch_bounds__(256) void scatter_kernel(const int*   __restrict__ esrc,
                                                      const int*   __restrict__ edst,
                                                      const float* __restrict__ eval,
                                                      const float* __restrict__ xw,
                                                      float*       __restrict__ out,
                                                      int E) {
    const int idx = blockIdx.x * 256 + threadIdx.x;   // < E*16 = 25.6M, fits int32
    const int e  = idx >> 4;
    const int c4 = (idx & 15) * 4;
    if (e >= E) return;
    const int   s = esrc[e];
    const int   d = edst[e];
    const float v = eval[e];
    const float4 g = *(const float4*)(xw + (size_t)s * OUT_DIM + c4);
    float* o = out + (size_t)d * OUT_DIM + c4;
    atomicAdd(o + 0, v * g.x);
    atomicAdd(o + 1, v * g.y);
    atomicAdd(o + 2, v * g.z);
    atomicAdd(o + 3, v * g.w);
}

// ---------------------------------------------------------------------------
// Kernel 4: in-place LeakyReLU (slope 0.01), float4 vectorized.
// ---------------------------------------------------------------------------
__global__ __launch_bounds__(256) void leaky_kernel(float4* __restrict__ out, int n4) {
    const int i = blockIdx.x * 256 + threadIdx.x;
    if (i < n4) {
        float4 v = out[i];
        v.x = v.x > 0.f ? v.x : 0.01f * v.x;
        v.y = v.y > 0.f ? v.y : 0.01f * v.y;
        v.z = v.z > 0.f ? v.z : 0.01f * v.z;
        v.w = v.w > 0.f ? v.w : 0.01f * v.w;
        out[i] = v;
    }
}

extern "C" void kernel_launch(void* const* d_in, const int* in_sizes, int n_in,
                              void* d_out, int out_size, void* d_ws, size_t ws_size,
                              hipStream_t stream) {
    const float* x    = (const float*)d_in[0];
    const float* W    = (const float*)d_in[1];
    const int*   esrc = (const int*)  d_in[2];
    const int*   edst = (const int*)  d_in[3];
    const float* eval = (const float*)d_in[4];
    float* out = (float*)d_out;
    float* xw  = (float*)d_ws;   // N*64 floats = 25.6 MB scratch

    const int N = in_sizes[0] / IN_DIM;
    const int E = in_sizes[2];

    // 1) dense projection via fp32 WMMA
    const int rowTiles = (N + 15) / 16;
    const int gemmBlks = (rowTiles + 7) / 8;
    gemm_xw_kernel<<<gemmBlks, 256, 0, stream>>>(x, W, xw, N);

    // 2) zero accumulation target (re-done every launch for graph replay)
    const int n  = N * OUT_DIM;
    const int n4 = n / 4;
    zero_kernel<<<(n4 + 255) / 256, 256, 0, stream>>>((float4*)out, n4);

    // 3) weighted scatter-add over edges (16 threads / edge, float4 gather)
    const long long work = (long long)E * 16;
    scatter_kernel<<<(int)((work + 255) / 256), 256, 0, stream>>>(esrc, edst, eval, xw, out, E);

    // 4) leaky_relu in place
    leaky_kernel<<<(n4 + 255) / 256, 256, 0, stream>>>((float4*)out, n4);
}